// GraphConvGRUCell_16801912062233
// MI455X (gfx1250) — compile-verified
//
#include <hip/hip_runtime.h>
#include <hip/hip_bf16.h>
#include <stdint.h>

typedef float v2f __attribute__((ext_vector_type(2)));
typedef float v8f __attribute__((ext_vector_type(8)));
typedef unsigned int u32x4 __attribute__((ext_vector_type(4)));
typedef int i32x4 __attribute__((ext_vector_type(4)));
typedef int i32x8 __attribute__((ext_vector_type(8)));

#define N_NODES   20000
#define IN_DIM    128
#define HID       128
#define COMB      256
#define N_EDGES   640000
#define NUM_KERN  2
#define K_DIFF    10

// ---------------------------------------------------------------------------
// TDM: DMA a [rows x cols] f32 tile (row stride rowStrideElts) into LDS.
// D# layout per CDNA5 ISA ch.8 (group0 128b, group1 256b); groups 2/3 unused
// (2D tensor). Issued by one wave; completion via TENSORcnt.
__device__ __forceinline__ void tdm_load_tile_f32(const float* gsrc, float* lds_dst,
                                                  int rows, int cols,
                                                  int rowStrideElts) {
  unsigned long long ga = (unsigned long long)(uintptr_t)gsrc;
  unsigned int lds = (unsigned int)(uintptr_t)lds_dst; // low 32b of generic = LDS offset
  u32x4 g0;
  g0[0] = 1u;                                     // count=1 (valid), no gather
  g0[1] = lds;                                    // lds_addr
  g0[2] = (unsigned int)(ga & 0xffffffffu);       // global_addr[31:0]
  g0[3] = (unsigned int)((ga >> 32) & 0x01ffffffu)// global_addr[56:32]
        | (2u << 30);                             // type=2 ("image")
  i32x8 g1;
  g1[0] = (2 << 16);                              // data_size=2 (4 bytes), no flags
  g1[1] = (cols & 0xffff) << 16;                  // tensor_dim0[15:0]
  g1[2] = ((cols >> 16) & 0xffff)                 // tensor_dim0[31:16]
        | ((rows & 0xffff) << 16);                // tensor_dim1[15:0]
  g1[3] = ((rows >> 16) & 0xffff)                 // tensor_dim1[31:16]
        | ((cols & 0xffff) << 16);                // tile_dim0
  g1[4] = rows & 0xffff;                          // tile_dim1 (tile_dim2=0)
  g1[5] = rowStrideElts;                          // tensor_dim0_stride[31:0]
  g1[6] = 0;                                      // stride hi / dim1_stride lo
  g1[7] = 0;
  i32x4 z4 = {0, 0, 0, 0};
#if defined(__clang_major__) && __clang_major__ >= 23
  i32x8 z8 = {0, 0, 0, 0, 0, 0, 0, 0};
  __builtin_amdgcn_tensor_load_to_lds(g0, g1, z4, z4, z8, 0);
#else
  __builtin_amdgcn_tensor_load_to_lds(g0, g1, z4, z4, 0);
#endif
}

// ---------------------------------------------------------------------------
// combined[i] = [ input[i] , (gate? gate[i]*hidden[i] : hidden[i]) ]
__global__ __launch_bounds__(256) void k_concat(const float* __restrict__ inp,
                                                const float* __restrict__ hid,
                                                const float* __restrict__ gate,
                                                float* __restrict__ out) {
  int t = blockIdx.x * blockDim.x + threadIdx.x;
  if (t >= N_NODES * (COMB / 4)) return;
  int row = t >> 6;
  int c   = t & 63;
  float4 v;
  if (c < 32) {
    v = *(const float4*)(inp + row * IN_DIM + c * 4);
  } else {
    int j = (c - 32) * 4;
    v = *(const float4*)(hid + row * HID + j);
    if (gate) {
      float4 g = *(const float4*)(gate + row * HID + j);
      v.x *= g.x; v.y *= g.y; v.z *= g.z; v.w *= g.w;
    }
  }
  *(float4*)(out + row * COMB + c * 4) = v;
}

__global__ __launch_bounds__(256) void k_zero(float4* __restrict__ p, int n4) {
  int t = blockIdx.x * blockDim.x + threadIdx.x;
  if (t < n4) p[t] = make_float4(0.f, 0.f, 0.f, 0.f);
}

__global__ __launch_bounds__(256) void k_bias(const float* __restrict__ b,
                                              float* __restrict__ acc) {
  int t = blockIdx.x * blockDim.x + threadIdx.x;
  if (t < N_NODES * HID) acc[t] = b[t & (HID - 1)];
}

// y[dst[e], :] += w[e] * x[src[e], :]   one wave per edge; edge data is
// wave-uniform -> force into SGPRs with readfirstlane (scalar loads).
__global__ __launch_bounds__(256) void k_spmm(const float* __restrict__ x,
                                              const int* __restrict__ src,
                                              const int* __restrict__ dst,
                                              const float* __restrict__ w,
                                              float* __restrict__ y) {
  int t    = blockIdx.x * blockDim.x + threadIdx.x;
  int e    = t >> 5;
  int lane = t & 31;
  if (e >= N_EDGES) return;
  int eu = __builtin_amdgcn_readfirstlane(e);      // provably uniform per wave
  int   s  = src[eu];
  int   d  = dst[eu];
  float we = __uint_as_float(
      __builtin_amdgcn_readfirstlane(__float_as_uint(w[eu])));
  const float4* xs = (const float4*)(x + (size_t)s * COMB) + lane * 2;
  float*        yd = y + (size_t)d * COMB + lane * 8;
  float4 v0 = xs[0];
  float4 v1 = xs[1];
  atomicAdd(yd + 0, we * v0.x);
  atomicAdd(yd + 1, we * v0.y);
  atomicAdd(yd + 2, we * v0.z);
  atomicAdd(yd + 3, we * v0.w);
  atomicAdd(yd + 4, we * v1.x);
  atomicAdd(yd + 5, we * v1.y);
  atomicAdd(yd + 6, we * v1.z);
  atomicAdd(yd + 7, we * v1.w);
}

// Chebyshev recurrence (in place): y = 2*y - x0
__global__ __launch_bounds__(256) void k_cheb(float4* __restrict__ y,
                                              const float4* __restrict__ x0) {
  int t = blockIdx.x * blockDim.x + threadIdx.x;
  if (t >= N_NODES * (COMB / 4)) return;
  float4 a = y[t];
  float4 b = x0[t];
  a.x = 2.f * a.x - b.x;  a.y = 2.f * a.y - b.y;
  a.z = 2.f * a.z - b.z;  a.w = 2.f * a.w - b.w;
  y[t] = a;
}

// acc(N,128) += X(N,256) @ Wm(256,128)  with V_WMMA_F32_16X16X4_F32.
// A tile (16x256, shared by all 8 waves) is DMA'd to LDS by the TDM once per
// block; waves read it via ds_load. 1 wave -> one 16x16 output tile.
__global__ __launch_bounds__(256) void k_gemm(const float* __restrict__ X,
                                              const float* __restrict__ Wm,
                                              float* __restrict__ acc) {
  __shared__ float sA[16 * COMB];                  // 16 KB of the 320 KB LDS
  const int lane    = threadIdx.x & 31;
  const int wave    = threadIdx.x >> 5;
  const int rowBase = blockIdx.x * 16;
  const int colBase = wave * 16;
  const int m  = lane & 15;    // A row / B,C,D col within tile
  const int hi = lane >> 4;    // half-wave select

  if (wave == 0) {             // one wave issues the tensor DMA for the block
    tdm_load_tile_f32(X + (size_t)rowBase * COMB, sA, 16, COMB, COMB);
    __builtin_amdgcn_s_wait_tensorcnt(0);
  }
  __syncthreads();

  // C/D 16x16 f32 layout: VGPR r -> M = 8*hi + r, N = colBase + m
  v8f c;
#pragma unroll
  for (int r = 0; r < 8; ++r)
    c[r] = acc[(size_t)(rowBase + 8 * hi + r) * HID + colBase + m];

  // A 16x4 layout: lane holds A[m][k0 + 2*hi .. +1]  (ds_load_b64, 8B aligned)
  const float* arow = sA + (size_t)m * COMB + 2 * hi;
  // B 4x16 layout: lane holds Wm[k0 + 2*hi][colBase + m] and the next K row
  const float* bcol = Wm + (size_t)(2 * hi) * HID + colBase + m;

  __builtin_prefetch(bcol + 64 * HID, 0, 0);       // global_prefetch_b8 on W

#pragma unroll
  for (int kk = 0; kk < COMB; kk += 4) {
    v2f a = *(const v2f*)(arow + kk);
    v2f b;
    b.x = bcol[(size_t)kk * HID];
    b.y = bcol[(size_t)kk * HID + HID];
    c = __builtin_amdgcn_wmma_f32_16x16x4_f32(
        /*neg_a=*/false, a, /*neg_b=*/false, b,
        /*c_mod=*/(short)0, c, /*reuse_a=*/false, /*reuse_b=*/false);
  }

#pragma unroll
  for (int r = 0; r < 8; ++r)
    acc[(size_t)(rowBase + 8 * hi + r) * HID + colBase + m] = c[r];
}

__global__ __launch_bounds__(256) void k_sigmoid(const float* __restrict__ a,
                                                 float* __restrict__ r) {
  int t = blockIdx.x * blockDim.x + threadIdx.x;
  if (t < N_NODES * HID) r[t] = 1.f / (1.f + __expf(-a[t]));
}

// out = u*h + (1-u)*sigmoid(accC)
__global__ __launch_bounds__(256) void k_final(const float* __restrict__ u,
                                               const float* __restrict__ hid,
                                               const float* __restrict__ accC,
                                               float* __restrict__ out) {
  int t = blockIdx.x * blockDim.x + threadIdx.x;
  if (t >= N_NODES * HID) return;
  float uu = u[t];
  float Cc = 1.f / (1.f + __expf(-accC[t]));
  out[t] = uu * hid[t] + (1.f - uu) * Cc;
}

// ---------------------------------------------------------------------------
extern "C" void kernel_launch(void* const* d_in, const int* in_sizes, int n_in,
                              void* d_out, int out_size, void* d_ws, size_t ws_size,
                              hipStream_t stream) {
  const float* inp  = (const float*)d_in[0];
  const float* hid  = (const float*)d_in[1];
  const int*   eidx = (const int*)  d_in[2];   // (2 kernels, 2, E)
  const float* ew   = (const float*)d_in[3];   // (2 kernels, E)
  const float* W    = (const float*)d_in[4];   // (21*256, 128)
  const float* bias = (const float*)d_in[5];   // (128,)
  float*       out  = (float*)d_out;

  float* ws = (float*)d_ws;
  const size_t FEAT = (size_t)N_NODES * COMB;
  const size_t HMAT = (size_t)N_NODES * HID;
  float* combined = ws;
  float* rot[3]   = {ws + FEAT, ws + 2 * FEAT, ws + 3 * FEAT};
  float* accG     = ws + 4 * FEAT;
  float* rbuf     = accG + HMAT;

  const int B = 256;
  const int gFeat4 = (N_NODES * (COMB / 4) + B - 1) / B;   // 5000
  const int gH     = (N_NODES * HID + B - 1) / B;          // 10000
  const int gSpmm  = (N_EDGES * 32 + B - 1) / B;           // 80000
  const int gGemm  = N_NODES / 16;                         // 1250

  auto gconv = [&](const float* xin, float* acc) {
    k_bias<<<gH, B, 0, stream>>>(bias, acc);
    k_gemm<<<gGemm, B, 0, stream>>>(xin, W, acc);          // m = 0 identity term
    int mIdx = 1;
    for (int k = 0; k < NUM_KERN; ++k) {
      const int*   src = eidx + (size_t)k * 2 * N_EDGES;
      const int*   dst = src + N_EDGES;
      const float* wk  = ew + (size_t)k * N_EDGES;

      int ip0 = -1, ip1 = 0;
      k_zero<<<gFeat4, B, 0, stream>>>((float4*)rot[ip1], N_NODES * (COMB / 4));
      k_spmm<<<gSpmm, B, 0, stream>>>(xin, src, dst, wk, rot[ip1]);
      k_gemm<<<gGemm, B, 0, stream>>>(rot[ip1], W + (size_t)mIdx * COMB * HID, acc);
      ++mIdx;

      const float* p0 = xin;
      const float* p1 = rot[ip1];
      for (int s = 2; s <= K_DIFF; ++s) {
        int inext = 0;
        for (int cand = 0; cand < 3; ++cand)
          if (cand != ip0 && cand != ip1) { inext = cand; break; }
        float* x2 = rot[inext];
        k_zero<<<gFeat4, B, 0, stream>>>((float4*)x2, N_NODES * (COMB / 4));
        k_spmm<<<gSpmm, B, 0, stream>>>(p1, src, dst, wk, x2);          // A x1
        k_cheb<<<gFeat4, B, 0, stream>>>((float4*)x2, (const float4*)p0); // 2*Ax1-x0
        k_gemm<<<gGemm, B, 0, stream>>>(x2, W + (size_t)mIdx * COMB * HID, acc);
        ++mIdx;
        ip0 = ip1; ip1 = inext;
        p0 = p1;   p1 = x2;
      }
    }
  };

  // combined = [input, hidden]
  k_concat<<<gFeat4, B, 0, stream>>>(inp, hid, nullptr, combined);

  // r = u = sigmoid(gconv(combined))   (reference computes the same gconv twice)
  gconv(combined, accG);
  k_sigmoid<<<gH, B, 0, stream>>>(accG, rbuf);

  // combined2 = [input, r*hidden]
  k_concat<<<gFeat4, B, 0, stream>>>(inp, hid, rbuf, combined);

  // C = sigmoid(gconv(combined2)); out = u*h + (1-u)*C
  gconv(combined, accG);
  k_final<<<gH, B, 0, stream>>>(rbuf, hid, accG, out);
}